// EPHGT_56942676410508
// MI455X (gfx1250) — compile-verified
//
#include <hip/hip_runtime.h>
#include <math.h>

// Problem constants (match reference)
#define NN 50000
#define EE 800000
#define HH 128
#define NHD 8
#define DD 16
#define TT 6
#define LL 2

typedef __attribute__((ext_vector_type(2))) float v2f;
typedef __attribute__((ext_vector_type(8))) float v8f;

#define LDSS 132  // padded LDS row stride (floats) to avoid bank conflicts

// ---------- helpers ----------
__device__ __forceinline__ float sigmoidf_(float x) { return 1.0f / (1.0f + __expf(-x)); }

__device__ __forceinline__ float geluf_(float x) {
  // tanh-approx GELU (jax.nn.gelu default)
  float x3 = x * x * x;
  return 0.5f * x * (1.0f + tanhf(0.7978845608028654f * (x + 0.044715f * x3)));
}

// order-preserving float<->uint map for atomicMax over signed floats
__device__ __forceinline__ unsigned f2ord(float f) {
  unsigned u = __float_as_uint(f);
  return (u & 0x80000000u) ? ~u : (u | 0x80000000u);
}
__device__ __forceinline__ float ord2f(unsigned u) {
  unsigned v = (u & 0x80000000u) ? (u & 0x7fffffffu) : ~u;
  return __uint_as_float(v);
}

// ---------- kernels ----------

// h = hidden + x_norm_abs @ W_pos[l]   (one thread per element)
__global__ void __launch_bounds__(256) pos_kernel(
    const float* __restrict__ hidden, const float* __restrict__ xna,
    const float* __restrict__ Wp /* [2,H] */, float* __restrict__ h) {
  int i = blockIdx.x * 256 + threadIdx.x;       // i < N*H (exact grid)
  int n = i >> 7, j = i & 127;
  h[i] = hidden[i] + xna[n * 2] * Wp[j] + xna[n * 2 + 1] * Wp[HH + j];
}

// out[n,:] = f(in[n,:]) @ W[type[n]]  with W = [T,H,H] row-major (k = input dim)
// 16-node tile per block; 8 waves, each wave owns one 16-wide output column tile.
// Computes all T types per tile (WMMA fp32 16x16x4), stores rows matching type.
__global__ void __launch_bounds__(256) proj_typed(
    const float* __restrict__ in, const float* __restrict__ W,
    const int* __restrict__ ntype, float* __restrict__ out, int applyGelu) {
  __shared__ float sA[16 * LDSS];
  __shared__ int sT[16];
  const int tile = blockIdx.x;
  const int tid = threadIdx.x;
  const int base_n = tile * 16;

  for (int idx = tid; idx < 16 * HH; idx += 256) {
    int r = idx >> 7, c = idx & 127;
    float v = in[(base_n + r) * HH + c];
    if (applyGelu) v = geluf_(v);
    sA[r * LDSS + c] = v;
  }
  if (tid < 16) sT[tid] = ntype[base_n + tid];
  __syncthreads();

  const int wave = tid >> 5;
  const int lane = tid & 31;
  const int half = lane >> 4;       // 0: K even pair, 1: K odd pair (per ISA A/B layout)
  const int m = lane & 15;          // A row / B column within tile
  const int colbase = wave * 16;
  const float* Arow = &sA[m * LDSS];

  for (int t = 0; t < TT; ++t) {
    const float* Wt = W + t * HH * HH;
    v8f c = {};
    for (int kk = 0; kk < HH / 4; ++kk) {
      int k0 = kk * 4 + 2 * half;
      v2f a, b;
      a.x = Arow[k0];
      a.y = Arow[k0 + 1];
      b.x = Wt[k0 * HH + colbase + m];
      b.y = Wt[(k0 + 1) * HH + colbase + m];
      c = __builtin_amdgcn_wmma_f32_16x16x4_f32(false, a, false, b, (short)0, c,
                                                false, false);
    }
    for (int r = 0; r < 8; ++r) {
      int mo = r + 8 * half;  // C/D layout: VGPR r -> M=r (lanes 0-15), M=r+8 (lanes 16-31)
      if (sT[mo] == t) out[(base_n + mo) * HH + colbase + m] = c[r];
    }
  }
}

// In-place per-head 16x16 transforms (one wave per (node-tile, head, which)):
//   which==0: q[n,h,:] <- (rel_att[h] @ q[n,h,:]) * rel_pri[h] * inv_sqrt_d
//   which==1: v[n,h,:] <- v[n,h,:] @ rel_msg[h]
__global__ void __launch_bounds__(32) head_xform(
    float* __restrict__ qbuf, float* __restrict__ vbuf,
    const float* __restrict__ relA, const float* __restrict__ relM,
    const float* __restrict__ relPri) {
  const int tile = blockIdx.x;
  const int head = blockIdx.y;
  const int which = blockIdx.z;
  const int lane = threadIdx.x;
  const int half = lane >> 4, m = lane & 15;
  float* buf = which ? vbuf : qbuf;
  const float* R = (which ? relM : relA) + head * DD * DD;
  const int nodebase = tile * 16;
  const int fb = head * DD;

  v8f c = {};
  for (int kk = 0; kk < 4; ++kk) {
    int k0 = kk * 4 + 2 * half;
    v2f a, b;
    a.x = buf[(nodebase + m) * HH + fb + k0];
    a.y = buf[(nodebase + m) * HH + fb + k0 + 1];
    if (which) {  // vm: B[k][f] = M[k*16+f]
      b.x = R[k0 * DD + m];
      b.y = R[(k0 + 1) * DD + m];
    } else {      // qr: out[n,d]=sum_f q[n,f]*R[d][f] -> B[f][d]=R[d*16+f]
      b.x = R[m * DD + k0];
      b.y = R[m * DD + k0 + 1];
    }
    c = __builtin_amdgcn_wmma_f32_16x16x4_f32(false, a, false, b, (short)0, c,
                                              false, false);
  }
  float scale = which ? 1.0f : (relPri[head] * 0.25f);  // inv_sqrt(16)=0.25
  for (int r = 0; r < 8; ++r) {
    int mo = r + 8 * half;
    buf[(nodebase + mo) * HH + fb + m] = c[r] * scale;
  }
}

// zero agg[N,H]; amax -> ordered(-inf)=0u ; den -> 0
__global__ void __launch_bounds__(256) init_kernel(
    float* __restrict__ agg, unsigned* __restrict__ amax, float* __restrict__ den) {
  int i = blockIdx.x * 256 + threadIdx.x;  // i < N*H (exact grid)
  agg[i] = 0.0f;
  if (i < NN * NHD) { amax[i] = 0u; den[i] = 0.0f; }
}

// att[e,h] = dot(k[src,h,:], qr[dst,h,:]); atomic segment-max per (dst,h)
__global__ void __launch_bounds__(256) edge_att(
    const float* __restrict__ kbuf, const float* __restrict__ qrbuf,
    const int* __restrict__ esrc, const int* __restrict__ edst,
    float* __restrict__ att, unsigned* __restrict__ amax) {
  int idx = blockIdx.x * 256 + threadIdx.x;  // idx < E*NH (exact grid)
  int e = idx >> 3, head = idx & 7;
  int s = esrc[e], d = edst[e];
  const float4* kv = (const float4*)(kbuf + s * HH + head * DD);
  const float4* qv = (const float4*)(qrbuf + d * HH + head * DD);
  float acc = 0.0f;
#pragma unroll
  for (int i = 0; i < 4; ++i) {
    float4 x = kv[i], y = qv[i];
    acc += x.x * y.x + x.y * y.y + x.z * y.z + x.w * y.w;
  }
  att[idx] = acc;
  atomicMax(&amax[d * NHD + head], f2ord(acc));
}

// ex = exp(att - max); atomic segment-sum of ex (in-place att->ex)
__global__ void __launch_bounds__(256) edge_exp(
    float* __restrict__ att, const unsigned* __restrict__ amax,
    float* __restrict__ den, const int* __restrict__ edst) {
  int idx = blockIdx.x * 256 + threadIdx.x;
  int e = idx >> 3, head = idx & 7;
  int d = edst[e];
  float mx = ord2f(amax[d * NHD + head]);
  float ex = __expf(att[idx] - mx);
  att[idx] = ex;
  atomicAdd(&den[d * NHD + head], ex);
}

// agg[dst,h,:] += alpha * vm[src,h,:]
__global__ void __launch_bounds__(256) edge_scatter(
    const float* __restrict__ ex, const float* __restrict__ den,
    const float* __restrict__ vm, const int* __restrict__ esrc,
    const int* __restrict__ edst, float* __restrict__ agg) {
  int idx = blockIdx.x * 256 + threadIdx.x;
  int e = idx >> 3, head = idx & 7;
  int s = esrc[e], d = edst[e];
  float alpha = ex[idx] / (den[d * NHD + head] + 1e-16f);
  const float4* vmv = (const float4*)(vm + s * HH + head * DD);
  float* ag = agg + d * HH + head * DD;
#pragma unroll
  for (int i = 0; i < 4; ++i) {
    float4 v = vmv[i];
    atomicAdd(&ag[i * 4 + 0], alpha * v.x);
    atomicAdd(&ag[i * 4 + 1], alpha * v.y);
    atomicAdd(&ag[i * 4 + 2], alpha * v.z);
    atomicAdd(&ag[i * 4 + 3], alpha * v.w);
  }
}

// skip-mix, priority, cn += hgt_out, hidden += tanh(cn).  One wave per node.
__global__ void __launch_bounds__(256) finalize_kernel(
    const float* __restrict__ outb, const float* __restrict__ hbuf,
    const int* __restrict__ ntype, const float* __restrict__ skipL,
    const float* __restrict__ wpriL, float* __restrict__ cn,
    float* __restrict__ hidden, float* __restrict__ pri) {
  int wave = threadIdx.x >> 5, lane = threadIdx.x & 31;
  int node = blockIdx.x * 8 + wave;  // N divisible by 8 -> exact
  float sk = sigmoidf_(skipL[ntype[node]]);
  const float4* o4 = (const float4*)(outb + node * HH);
  const float4* h4 = (const float4*)(hbuf + node * HH);
  const float4* w4 = (const float4*)wpriL;
  float4* c4 = (float4*)(cn + node * HH);
  float4* hi4 = (float4*)(hidden + node * HH);

  float4 o = o4[lane], hh = h4[lane], wp = w4[lane], cc = c4[lane], hv = hi4[lane];
  float4 ho;
  ho.x = sk * o.x + (1.0f - sk) * hh.x;
  ho.y = sk * o.y + (1.0f - sk) * hh.y;
  ho.z = sk * o.z + (1.0f - sk) * hh.z;
  ho.w = sk * o.w + (1.0f - sk) * hh.w;
  float p = ho.x * wp.x + ho.y * wp.y + ho.z * wp.z + ho.w * wp.w;
  cc.x += ho.x; cc.y += ho.y; cc.z += ho.z; cc.w += ho.w;
  hv.x += tanhf(cc.x); hv.y += tanhf(cc.y); hv.z += tanhf(cc.z); hv.w += tanhf(cc.w);
  c4[lane] = cc;
  hi4[lane] = hv;
  for (int off = 16; off; off >>= 1) p += __shfl_xor(p, off, 32);
  if (lane == 0) pri[node] = sigmoidf_(p);
}

// ---------- launch ----------
extern "C" void kernel_launch(void* const* d_in, const int* in_sizes, int n_in,
                              void* d_out, int out_size, void* d_ws, size_t ws_size,
                              hipStream_t stream) {
  (void)in_sizes; (void)n_in; (void)out_size; (void)ws_size;

  const float* in_hidden = (const float*)d_in[0];
  const float* in_cn     = (const float*)d_in[1];
  const float* xna       = (const float*)d_in[2];
  const int*   ntype     = (const int*)d_in[3];
  const int*   esrc      = (const int*)d_in[4];
  const int*   edst      = (const int*)d_in[5];
  const float* Wpos      = (const float*)d_in[6];   // [L,2,H]
  const float* Wk        = (const float*)d_in[7];   // [L,T,H,H]
  const float* Wq        = (const float*)d_in[8];
  const float* Wv        = (const float*)d_in[9];
  const float* Wa        = (const float*)d_in[10];
  const float* relA      = (const float*)d_in[11];  // [L,NH,D,D]
  const float* relM      = (const float*)d_in[12];
  const float* relP      = (const float*)d_in[13];  // [L,NH]
  const float* skip      = (const float*)d_in[14];  // [L,T]
  const float* wpri      = (const float*)d_in[15];  // [L,H]

  float* hidden = (float*)d_out;            // [N,H]
  float* cn     = hidden + (size_t)NN * HH; // [N,H]
  float* pri    = cn + (size_t)NN * HH;     // [N]

  const size_t NHID = (size_t)NN * HH;  // 6.4M floats (== E*NH)
  float* ws   = (float*)d_ws;
  float* hb   = ws;             // h = hidden + pos
  float* kb   = hb + NHID;      // k
  float* qb   = kb + NHID;      // q -> qr (in-place)
  float* vb   = qb + NHID;      // v -> vm (in-place)
  float* agg  = vb + NHID;      // aggregated messages [N,H]
  float* outb = agg + NHID;     // Wa projection output [N,H]
  float* att  = outb + NHID;    // att -> ex, [E,NH]
  unsigned* amax = (unsigned*)(att + (size_t)EE * NHD);  // [N,NH]
  float* den  = (float*)(amax + (size_t)NN * NHD);       // [N,NH]

  hipMemcpyAsync(hidden, in_hidden, NHID * sizeof(float), hipMemcpyDeviceToDevice, stream);
  hipMemcpyAsync(cn, in_cn, NHID * sizeof(float), hipMemcpyDeviceToDevice, stream);

  const int NTILE = NN / 16;          // 3125 exact
  const int ELEM_BLKS = NN * HH / 256;    // 25000 exact
  const int EDGE_BLKS = EE * NHD / 256;   // 25000 exact

  for (int l = 0; l < LL; ++l) {
    const int wOff  = l * TT * HH * HH;
    const int rOff  = l * NHD * DD * DD;

    pos_kernel<<<ELEM_BLKS, 256, 0, stream>>>(hidden, xna, Wpos + l * 2 * HH, hb);

    proj_typed<<<NTILE, 256, 0, stream>>>(hb, Wk + wOff, ntype, kb, 0);
    proj_typed<<<NTILE, 256, 0, stream>>>(hb, Wq + wOff, ntype, qb, 0);
    proj_typed<<<NTILE, 256, 0, stream>>>(hb, Wv + wOff, ntype, vb, 0);

    head_xform<<<dim3(NTILE, NHD, 2), 32, 0, stream>>>(qb, vb, relA + rOff,
                                                       relM + rOff, relP + l * NHD);

    init_kernel<<<ELEM_BLKS, 256, 0, stream>>>(agg, amax, den);
    edge_att<<<EDGE_BLKS, 256, 0, stream>>>(kb, qb, esrc, edst, att, amax);
    edge_exp<<<EDGE_BLKS, 256, 0, stream>>>(att, amax, den, edst);
    edge_scatter<<<EDGE_BLKS, 256, 0, stream>>>(att, den, vb, esrc, edst, agg);

    proj_typed<<<NTILE, 256, 0, stream>>>(agg, Wa + wOff, ntype, outb, 1);

    finalize_kernel<<<NN / 8, 256, 0, stream>>>(outb, hb, ntype, skip + l * TT,
                                                wpri + l * HH, cn, hidden, pri);
  }
}